// RS_KGA0_att2_subexp1_69002944577611
// MI455X (gfx1250) — compile-verified
//
#include <hip/hip_runtime.h>

#define N_USERS 50000
#define N_ITEMS 30000
#define N_ENT   100000
#define N_REL   8
#define E_KG    250000
#define E_UI    1000000
#define DIMF    128
#define MEAN_ROWS_PER_BLOCK 500

typedef __attribute__((ext_vector_type(2))) float v2f;
typedef __attribute__((ext_vector_type(8))) float v8f;

// ---------------- counting / inversion ----------------

__global__ __launch_bounds__(256) void k_count_kg(const int* __restrict__ kg_dst,
                                                  float* __restrict__ cnt) {
  int e = blockIdx.x * 256 + threadIdx.x;
  int r = blockIdx.y;
  if (e < E_KG) {
    int d = kg_dst[(size_t)r * E_KG + e];
    atomicAdd(&cnt[(size_t)r * N_ENT + d], 1.0f);
  }
}

__global__ __launch_bounds__(256) void k_count_ui(const int* __restrict__ ui_dst,
                                                  float* __restrict__ cnt) {
  int e = blockIdx.x * 256 + threadIdx.x;
  if (e < E_UI) atomicAdd(&cnt[ui_dst[e]], 1.0f);
}

__global__ __launch_bounds__(256) void k_invert(float* __restrict__ x, int n) {
  int i = blockIdx.x * 256 + threadIdx.x;
  if (i < n) x[i] = 1.0f / fmaxf(x[i], 1.0f);
}

// ---------------- hop 1: per-relation 16-dim slice seg-mean ----------------
// thread = (r, e, c in 0..15); contributions pre-scaled by 1/count(dst)

__global__ __launch_bounds__(256) void k_hop1_scatter(const float* __restrict__ all_embed,
                                                      const int* __restrict__ kg_src,
                                                      const int* __restrict__ kg_dst,
                                                      const float* __restrict__ inv_kg,
                                                      float* __restrict__ ent0) {
  int tid = blockIdx.x * 256 + threadIdx.x;
  int r = blockIdx.y;
  int e = tid >> 4;
  int c = tid & 15;
  int s = kg_src[(size_t)r * E_KG + e];
  int d = kg_dst[(size_t)r * E_KG + e];
  float inv = inv_kg[(size_t)r * N_ENT + d];
  int col = r * 16 + c;
  float v = all_embed[(size_t)s * DIMF + col] * inv;
  atomicAdd(&ent0[(size_t)d * DIMF + col], v);
}

// ---------------- hops 2..3: full 128-dim seg-mean, summed over relations ----
// wave per edge: 32 lanes x float4 = coalesced 512B row, pre-scaled atomics

__global__ __launch_bounds__(256) void k_hop_scatter(const float* __restrict__ ent0,
                                                     const int* __restrict__ kg_src,
                                                     const int* __restrict__ kg_dst,
                                                     const float* __restrict__ inv_kg,
                                                     float* __restrict__ entN) {
  int tid = blockIdx.x * 256 + threadIdx.x;
  int r = blockIdx.y;
  int e = tid >> 5;
  int c4 = tid & 31;
  int s = kg_src[(size_t)r * E_KG + e];
  int d = kg_dst[(size_t)r * E_KG + e];
  float inv = inv_kg[(size_t)r * N_ENT + d];
  float4 v = reinterpret_cast<const float4*>(ent0 + (size_t)s * DIMF)[c4];
  float* dst = entN + (size_t)d * DIMF + c4 * 4;
  atomicAdd(dst + 0, v.x * inv);
  atomicAdd(dst + 1, v.y * inv);
  atomicAdd(dst + 2, v.z * inv);
  atomicAdd(dst + 3, v.w * inv);
}

__global__ __launch_bounds__(256) void k_copy4(const float* __restrict__ src,
                                               float* __restrict__ dst, int n4) {
  int i = blockIdx.x * 256 + threadIdx.x;
  if (i < n4) reinterpret_cast<float4*>(dst)[i] = reinterpret_cast<const float4*>(src)[i];
}

__global__ __launch_bounds__(256) void k_hop_finish(const float* __restrict__ entN,
                                                    float* __restrict__ ent_res,
                                                    float* __restrict__ ent0, int n4) {
  int i = blockIdx.x * 256 + threadIdx.x;
  if (i < n4) {
    float4 v = reinterpret_cast<const float4*>(entN)[i];
    float4 a = reinterpret_cast<float4*>(ent_res)[i];
    a.x += v.x; a.y += v.y; a.z += v.z; a.w += v.w;
    reinterpret_cast<float4*>(ent_res)[i] = a;
    reinterpret_cast<float4*>(ent0)[i] = v;
  }
}

// ---------------- user-item aggregation (two payloads per edge) -------------

__global__ __launch_bounds__(256) void k_ui_scatter(const float* __restrict__ ent_res,
                                                    const float* __restrict__ all_embed_cf,
                                                    const int* __restrict__ ui_src_item,
                                                    const int* __restrict__ ui_dst_user,
                                                    const float* __restrict__ inv_ui,
                                                    float* __restrict__ user_f,
                                                    float* __restrict__ user_f0) {
  int tid = blockIdx.x * 256 + threadIdx.x;
  int e = tid >> 5;
  int c4 = tid & 31;
  int it = ui_src_item[e];
  int u  = ui_dst_user[e];
  float inv = inv_ui[u];
  float4 a = reinterpret_cast<const float4*>(ent_res + (size_t)it * DIMF)[c4];
  float4 b = reinterpret_cast<const float4*>(all_embed_cf + (size_t)(N_USERS + it) * DIMF)[c4];
  float* pf  = user_f  + (size_t)u * DIMF + c4 * 4;
  float* pf0 = user_f0 + (size_t)u * DIMF + c4 * 4;
  atomicAdd(pf  + 0, a.x * inv); atomicAdd(pf  + 1, a.y * inv);
  atomicAdd(pf  + 2, a.z * inv); atomicAdd(pf  + 3, a.w * inv);
  atomicAdd(pf0 + 0, b.x * inv); atomicAdd(pf0 + 1, b.y * inv);
  atomicAdd(pf0 + 2, b.z * inv); atomicAdd(pf0 + 3, b.w * inv);
}

// ---------------- item_mean via V_WMMA_F32_16X16X4_F32 (exact f32) ----------
// ones(16x4) @ B(4x16)  ->  every row of D = column sums of the 4-row chunk.
// Each wave owns one 16-column tile; block of 8 waves covers all 128 columns.
// B layout (f32, 4x16): lanes 0-15 = N, V0/V1 = K0/K1; lanes 16-31 = N, K2/K3.

__global__ __launch_bounds__(256) void k_item_mean(const float* __restrict__ ent_res,
                                                   float* __restrict__ item_mean) {
  int wave = threadIdx.x >> 5;       // 0..7 -> column tile
  int lane = threadIdx.x & 31;
  int colBase = wave * 16;
  int n = lane & 15;
  int khalf = (lane >> 4) * 2;       // rows +0/+1 (low half) or +2/+3 (high half)
  int rowBase = blockIdx.x * MEAN_ROWS_PER_BLOCK;

  v2f a; a[0] = 1.0f; a[1] = 1.0f;   // all-ones A matrix
  v8f acc = {};
  for (int r0 = 0; r0 < MEAN_ROWS_PER_BLOCK; r0 += 4) {
    const float* p = ent_res + (size_t)(rowBase + r0 + khalf) * DIMF + colBase + n;
    v2f b; b[0] = p[0]; b[1] = p[DIMF];
    acc = __builtin_amdgcn_wmma_f32_16x16x4_f32(false, a, false, b,
                                                (short)0, acc, false, false);
  }
  // D rows identical; VGPR0 of lanes 0-15 holds (M=0, N=lane)
  if (lane < 16) atomicAdd(&item_mean[colBase + lane], acc[0]);
}

// ---------------- per-user attention + output ------------------------------

__global__ __launch_bounds__(128) void k_user_attn(const float* __restrict__ user_f,
                                                   const float* __restrict__ user_f0,
                                                   const float* __restrict__ all_embed_cf,
                                                   const float* __restrict__ item_mean,
                                                   float* __restrict__ user_out) {
  int u = blockIdx.x;
  int c = threadIdx.x;                       // 0..127, relation = c/16
  float uf  = user_f[(size_t)u * DIMF + c];
  float ucf = all_embed_cf[(size_t)u * DIMF + c];  // user_cf
  float im  = item_mean[c] * (1.0f / 30000.0f);
  float pu = uf * ucf;
  float pm = im * ucf;
  // sum within each aligned 16-lane group (one relation) — wave32
  #pragma unroll
  for (int off = 8; off >= 1; off >>= 1) {
    pu += __shfl_xor(pu, off, 32);
    pm += __shfl_xor(pm, off, 32);
  }
  float ua = fmaxf(pu, 0.0f) + 1e-10f;
  float ma = fmaxf(pm, 0.0f) + 1e-8f;
  float att = fmaxf(ua / ma - 1.0f, 0.0f) + 0.01f;   // K_ATT = 1.0
  float score = tanhf(att);
  user_out[(size_t)u * DIMF + c] = score * uf + user_f0[(size_t)u * DIMF + c] + ucf;
}

__global__ __launch_bounds__(256) void k_item_out(const float* __restrict__ ent_res,
                                                  const float* __restrict__ all_embed_cf,
                                                  float* __restrict__ out, int n4) {
  int i = blockIdx.x * 256 + threadIdx.x;
  if (i < n4) {
    float4 a = reinterpret_cast<const float4*>(ent_res)[i];
    float4 b = reinterpret_cast<const float4*>(all_embed_cf + (size_t)N_USERS * DIMF)[i];
    a.x += b.x; a.y += b.y; a.z += b.z; a.w += b.w;
    reinterpret_cast<float4*>(out)[i] = a;
  }
}

// ---------------------------------------------------------------------------

extern "C" void kernel_launch(void* const* d_in, const int* in_sizes, int n_in,
                              void* d_out, int out_size, void* d_ws, size_t ws_size,
                              hipStream_t stream) {
  const float* all_embed    = (const float*)d_in[0];
  const float* all_embed_cf = (const float*)d_in[1];
  const int*   kg_src       = (const int*)d_in[2];
  const int*   kg_dst       = (const int*)d_in[3];
  const int*   ui_src       = (const int*)d_in[4];
  const int*   ui_dst       = (const int*)d_in[5];
  float* out = (float*)d_out;

  // workspace layout (floats): ~208 MB total
  float* ws        = (float*)d_ws;
  float* ent0      = ws;                                   // N_ENT*128
  float* entN      = ent0 + (size_t)N_ENT * DIMF;          // N_ENT*128
  float* ent_res   = entN + (size_t)N_ENT * DIMF;          // N_ENT*128
  float* inv_kg    = ent_res + (size_t)N_ENT * DIMF;       // N_REL*N_ENT
  float* user_f    = inv_kg + (size_t)N_REL * N_ENT;       // N_USERS*128
  float* user_f0   = user_f + (size_t)N_USERS * DIMF;      // N_USERS*128
  float* inv_ui    = user_f0 + (size_t)N_USERS * DIMF;     // N_USERS
  float* item_mean = inv_ui + N_USERS;                     // 128

  // zero accumulators (ent_res is fully written by k_copy4)
  hipMemsetAsync(ent0, 0, (size_t)2 * N_ENT * DIMF * sizeof(float), stream);
  hipMemsetAsync(inv_kg, 0,
                 ((size_t)N_REL * N_ENT + 2 * (size_t)N_USERS * DIMF + N_USERS + DIMF)
                   * sizeof(float), stream);

  dim3 b256(256);
  const int n4 = N_ENT * DIMF / 4;

  // per-relation destination counts -> inverse (hop-invariant)
  k_count_kg<<<dim3((E_KG + 255) / 256, N_REL), b256, 0, stream>>>(kg_dst, inv_kg);
  k_invert<<<(N_REL * N_ENT + 255) / 256, b256, 0, stream>>>(inv_kg, N_REL * N_ENT);

  // hop 1 (16-dim slices) and ent_res init
  k_hop1_scatter<<<dim3(E_KG * 16 / 256, N_REL), b256, 0, stream>>>(
      all_embed, kg_src, kg_dst, inv_kg, ent0);
  k_copy4<<<(n4 + 255) / 256, b256, 0, stream>>>(ent0, ent_res, n4);

  // hops 2..3 (full 128-dim)
  for (int h = 0; h < 2; ++h) {
    k_hop_scatter<<<dim3(E_KG * 32 / 256, N_REL), b256, 0, stream>>>(
        ent0, kg_src, kg_dst, inv_kg, entN);
    k_hop_finish<<<(n4 + 255) / 256, b256, 0, stream>>>(entN, ent_res, ent0, n4);
    if (h == 0) hipMemsetAsync(entN, 0, (size_t)N_ENT * DIMF * sizeof(float), stream);
  }

  // user-item aggregation
  k_count_ui<<<(E_UI + 255) / 256, b256, 0, stream>>>(ui_dst, inv_ui);
  k_invert<<<(N_USERS + 255) / 256, b256, 0, stream>>>(inv_ui, N_USERS);
  k_ui_scatter<<<E_UI * 32 / 256, b256, 0, stream>>>(
      ent_res, all_embed_cf, ui_src, ui_dst, inv_ui, user_f, user_f0);

  // item_mean (WMMA f32 reduction), then attention + outputs
  k_item_mean<<<N_ITEMS / MEAN_ROWS_PER_BLOCK, b256, 0, stream>>>(ent_res, item_mean);
  k_user_attn<<<N_USERS, 128, 0, stream>>>(user_f, user_f0, all_embed_cf, item_mean,
                                           out + (size_t)N_ITEMS * DIMF);
  k_item_out<<<(N_ITEMS * DIMF / 4 + 255) / 256, b256, 0, stream>>>(
      ent_res, all_embed_cf, out, N_ITEMS * DIMF / 4);
}